// Net_73770358276923
// MI455X (gfx1250) — compile-verified
//
#include <hip/hip_runtime.h>
#include <math.h>

typedef __attribute__((ext_vector_type(2))) float v2f;
typedef __attribute__((ext_vector_type(8))) float v8f;

#define KGRID 5

// ---------------- device helpers ----------------

__device__ __forceinline__ float pseudo_coord(float d, float m) {
  float mm = fmaxf(m, 1e-12f);
  float v = d / (2.0f * mm) + 0.5f;
  return fminf(fmaxf(v, 0.0f), 1.0f);
}

__device__ __forceinline__ void bspline2(float px, float py, float bw[4], int kx[4]) {
  float v0 = px * (float)(KGRID - 1), v1 = py * (float)(KGRID - 1);
  float l0 = fminf(fmaxf(floorf(v0), 0.0f), (float)(KGRID - 2));
  float l1 = fminf(fmaxf(floorf(v1), 0.0f), (float)(KGRID - 2));
  float f0 = v0 - l0, f1 = v1 - l1;
  int k0 = (int)l0, k1 = (int)l1;
  bw[0] = (1.0f - f0) * (1.0f - f1);
  bw[1] = (1.0f - f0) * f1;
  bw[2] = f0 * (1.0f - f1);
  bw[3] = f0 * f1;
  kx[0] = k0 * KGRID + k1;
  kx[1] = k0 * KGRID + k1 + 1;
  kx[2] = (k0 + 1) * KGRID + k1;
  kx[3] = (k0 + 1) * KGRID + k1 + 1;
}

// ---------------- utility kernels ----------------

__global__ void zero_kernel(float* p, size_t n) {
  size_t i = (size_t)blockIdx.x * blockDim.x + threadIdx.x;
  size_t stride = (size_t)gridDim.x * blockDim.x;
  for (; i < n; i += stride) p[i] = 0.0f;
}

// max over explicit edge list of |pos[col]-pos[row]| (emask==1 everywhere)
__global__ void absmax_edges_kernel(const float* pos, const int* row, const int* col,
                                    int E, float* m) {
  int e = blockIdx.x * blockDim.x + threadIdx.x;
  float v = 0.0f;
  if (e < E) {
    int r = row[e], c = col[e];
    float dx = pos[2 * c] - pos[2 * r];
    float dy = pos[2 * c + 1] - pos[2 * r + 1];
    v = fmaxf(fabsf(dx), fabsf(dy));
  }
  for (int off = 16; off > 0; off >>= 1) v = fmaxf(v, __shfl_xor(v, off, 32));
  if ((threadIdx.x & 31) == 0) atomicMax((int*)m, __float_as_int(v));
}

// max over dense BxPxP pair grid with exist mask and lr!=lc
__global__ void absmax_dense_kernel(const float* pos, const int* exist, int P,
                                    int nPairs, float* m) {
  int e = blockIdx.x * blockDim.x + threadIdx.x;
  float v = 0.0f;
  if (e < nPairs) {
    int P2 = P * P;
    int g = e / P2, lr = (e / P) % P, lc = e % P;
    if (exist[e] != 0 && lr != lc) {
      int r = g * P + lr, c = g * P + lc;
      float dx = pos[2 * c] - pos[2 * r];
      float dy = pos[2 * c + 1] - pos[2 * r + 1];
      v = fmaxf(fabsf(dx), fabsf(dy));
    }
  }
  for (int off = 16; off > 0; off >>= 1) v = fmaxf(v, __shfl_xor(v, off, 32));
  if ((threadIdx.x & 31) == 0) atomicMax((int*)m, __float_as_int(v));
}

// ---------------- layer 1 (in_feats == 1) ----------------

// wave per edge, lane = output feature (F=32)
__global__ void spline_edge_l1_kernel(const float* x, const float* pos, const int* row,
                                      const int* col, const float* W1, const float* mptr,
                                      float* agg, float* deg, int E) {
  int t = blockIdx.x * blockDim.x + threadIdx.x;
  int e = t >> 5, lane = t & 31;
  if (e >= E) return;
  int r = row[e], c = col[e];
  float m = *mptr;
  float px = pseudo_coord(pos[2 * c] - pos[2 * r], m);
  float py = pseudo_coord(pos[2 * c + 1] - pos[2 * r + 1], m);
  float bw[4]; int kx[4];
  bspline2(px, py, bw, kx);
  float xc = x[c];
  float acc = 0.0f;
#pragma unroll
  for (int s = 0; s < 4; ++s) acc += bw[s] * W1[kx[s] * 32 + lane];
  atomicAdd(&agg[(size_t)r * 32 + lane], xc * acc);
  if (lane == 0) atomicAdd(&deg[r], 1.0f);
}

__global__ void node_fin_l1_kernel(float* agg, const float* deg, const float* x,
                                   const float* R1, int nNodes) {
  int t = blockIdx.x * blockDim.x + threadIdx.x;
  int n = t >> 5, lane = t & 31;
  if (n >= nNodes) return;
  float d = fmaxf(deg[n], 1.0f);
  float v = agg[(size_t)n * 32 + lane] / d + x[n] * R1[lane];
  agg[(size_t)n * 32 + lane] = fmaxf(v, 0.0f);
}

// ---------------- voxel pooling ----------------

// wave per node, lanes cover F features; h >= 0 so int-bitcast atomicMax works
__global__ void pool_nodes_kernel(const float* h, int F, const float* pos, int npg, int G,
                                  int* nodecl, float* cnt, float* possum, float* feat,
                                  int nNodes) {
  int t = blockIdx.x * blockDim.x + threadIdx.x;
  int n = t >> 5, lane = t & 31;
  if (n >= nNodes) return;
  float px = pos[2 * n], py = pos[2 * n + 1];
  int cx = min(max((int)(px * (float)G), 0), G - 1);
  int cy = min(max((int)(py * (float)G), 0), G - 1);
  int P = G * G;
  int cl = (n / npg) * P + cx * G + cy;
  if (lane == 0) {
    nodecl[n] = cl;
    atomicAdd(&cnt[cl], 1.0f);
    atomicAdd(&possum[2 * cl], px);
    atomicAdd(&possum[2 * cl + 1], py);
  }
  for (int o = lane; o < F; o += 32)
    atomicMax((int*)&feat[(size_t)cl * F + o], __float_as_int(h[(size_t)n * F + o]));
}

__global__ void pool_div_kernel(float* possum, const float* cnt, int C) {
  int c = blockIdx.x * blockDim.x + threadIdx.x;
  if (c >= C) return;
  float d = fmaxf(cnt[c], 1.0f);
  possum[2 * c] /= d;
  possum[2 * c + 1] /= d;
}

// build exist mask for pooled graph from explicit edge list (layer-1 edges, emask==1)
__global__ void pool_edges_l1_kernel(const int* row, const int* col, const int* nodecl,
                                     int P, int* exist, int E) {
  int e = blockIdx.x * blockDim.x + threadIdx.x;
  if (e >= E) return;
  int cr = nodecl[row[e]], cc = nodecl[col[e]];
  int g = cr / P, lr = cr % P, lc = cc % P;
  exist[g * P * P + lr * P + lc] = 1;  // idempotent racy store
}

// build exist mask for pooled graph from dense masked pair grid
__global__ void pool_edges_dense_kernel(const int* existOld, int Pold, const int* noclNew,
                                        int Pnew, int* existNew, int nPairsOld) {
  int e = blockIdx.x * blockDim.x + threadIdx.x;
  if (e >= nPairsOld) return;
  int P2 = Pold * Pold;
  int g = e / P2, lr = (e / Pold) % Pold, lc = e % Pold;
  if (existOld[e] == 0 || lr == lc) return;
  int r = g * Pold + lr, c = g * Pold + lc;
  int cr = noclNew[r], cc = noclNew[c];
  int gn = cr / Pnew;
  existNew[gn * Pnew * Pnew + (cr % Pnew) * Pnew + (cc % Pnew)] = 1;
}

// ---------------- WMMA fp32 GEMM: C = A(MxK) * B(KxN), all row-major ----------------
// Requires M%16==0, N%16==0, K%4==0. One wave -> a 16x(16*JT) C strip (JT accumulators)
// via V_WMMA_F32_16X16X4_F32. JT is a compile-time template parameter so the inner
// body is straight-line: one 64-bit A load + 2*JT B loads form a clause, then JT
// independent WMMAs drain with decreasing s_wait_loadcnt (A fragment reused JT times).
// Only wave-level early-exit remains, so EXEC is all-ones at every WMMA (ISA 7.12).

template <int JT>
__global__ void wmma_gemm_kernel(const float* __restrict__ A, int lda,
                                 const float* __restrict__ Bm, int ldb,
                                 float* __restrict__ C, int ldc,
                                 int M, int K, int nblk, int jt0) {
  int t = blockIdx.x * blockDim.x + threadIdx.x;
  int wave = t >> 5;
  int lane = t & 31;
  int nwaves = (M >> 4) * nblk;
  if (wave >= nwaves) return;           // wave-uniform early exit
  int mt = wave / nblk, nb = wave % nblk;
  int m0 = mt << 4;
  int n0base = (jt0 + nb * JT) << 4;
  int hl = lane >> 4;                   // half-wave select
  int l16 = lane & 15;
  v8f zero = {0.f, 0.f, 0.f, 0.f, 0.f, 0.f, 0.f, 0.f};
  v8f acc[JT];
#pragma unroll
  for (int j = 0; j < JT; ++j) acc[j] = zero;
  const float* arow = A + (size_t)(m0 + l16) * lda;
  for (int kk = 0; kk < K; kk += 4) {
    // A 16x4 tile: VGPR0 -> K = 2*hl, VGPR1 -> K = 2*hl+1, M = l16
    v2f a;
    a.x = arow[kk + 2 * hl];
    a.y = arow[kk + 2 * hl + 1];
    const float* b0 = Bm + (size_t)(kk + 2 * hl) * ldb + n0base + l16;
    const float* b1 = b0 + ldb;
    v2f bfrag[JT];
#pragma unroll
    for (int j = 0; j < JT; ++j) {      // all B loads issue as one clause
      bfrag[j].x = b0[j * 16];
      bfrag[j].y = b1[j * 16];
    }
#pragma unroll
    for (int j = 0; j < JT; ++j)        // JT independent WMMAs share the A fragment
      acc[j] = __builtin_amdgcn_wmma_f32_16x16x4_f32(false, a, false, bfrag[j],
                                                     (short)0, acc[j], false, false);
  }
#pragma unroll
  for (int j = 0; j < JT; ++j) {
    int n0 = n0base + j * 16;
#pragma unroll
    for (int r = 0; r < 8; ++r)
      C[(size_t)(m0 + r + 8 * hl) * ldc + n0 + l16] = acc[j][r];
  }
}

// repack W (Kspline, Kin, 64) -> row-major Bmat (Kin, Kspline*64)
__global__ void repack_w_kernel(const float* W, int Kin, float* Bm, int total) {
  int t = blockIdx.x * blockDim.x + threadIdx.x;
  if (t >= total) return;
  int i = t / 1600;
  int j = t % 1600;
  int k = j >> 6, o = j & 63;
  Bm[t] = W[((size_t)k * Kin + i) * 64 + o];
}

// ---------------- dense-pair spline conv (layers 2 & 3, F = 64) ----------------

__global__ void spline_edge_dense_kernel(const float* __restrict__ XW, const float* pos,
                                         const int* exist, const float* mptr, int P,
                                         float* agg, float* deg, int nPairs) {
  int t = blockIdx.x * blockDim.x + threadIdx.x;
  int e = t >> 5, lane = t & 31;
  if (e >= nPairs) return;
  int P2 = P * P;
  int g = e / P2, lr = (e / P) % P, lc = e % P;
  if (lr == lc || exist[e] == 0) return;
  int r = g * P + lr, c = g * P + lc;
  float m = *mptr;
  float px = pseudo_coord(pos[2 * c] - pos[2 * r], m);
  float py = pseudo_coord(pos[2 * c + 1] - pos[2 * r + 1], m);
  float bw[4]; int kx[4];
  bspline2(px, py, bw, kx);
  const float* base = XW + (size_t)c * 1600;
#pragma unroll
  for (int u = 0; u < 2; ++u) {
    int o = lane + u * 32;
    float acc = 0.0f;
#pragma unroll
    for (int s = 0; s < 4; ++s) acc += bw[s] * base[kx[s] * 64 + o];
    atomicAdd(&agg[(size_t)r * 64 + o], acc);
  }
  if (lane == 0) atomicAdd(&deg[r], 1.0f);
}

__global__ void node_fin_kernel(float* agg, const float* deg, const float* root, int F,
                                int nNodes) {
  int t = blockIdx.x * blockDim.x + threadIdx.x;
  int n = t >> 5, lane = t & 31;
  if (n >= nNodes) return;
  float d = fmaxf(deg[n], 1.0f);
  for (int o = lane; o < F; o += 32) {
    float v = agg[(size_t)n * F + o] / d + root[(size_t)n * F + o];
    agg[(size_t)n * F + o] = fmaxf(v, 0.0f);
  }
}

// ---------------- final 2x2 pool + FC head ----------------

__global__ void final_pool_kernel(const float* h, const float* pos, int nNodes, int npg,
                                  float* feat4) {
  int t = blockIdx.x * blockDim.x + threadIdx.x;
  int n = t >> 5, lane = t & 31;
  if (n >= nNodes) return;
  int g = n / npg;
  int cx = min(max((int)floorf(pos[2 * n] + 0.5f), 0), 1);
  int cy = min(max((int)floorf(pos[2 * n + 1] + 0.5f), 0), 1);
  int cl = g * 4 + cx * 2 + cy;
  for (int o = lane; o < 64; o += 32)
    atomicMax((int*)&feat4[(size_t)cl * 64 + o], __float_as_int(h[(size_t)n * 64 + o]));
}

__global__ void elu_bias_kernel(float* z, const float* b, int total, int cols) {
  int t = blockIdx.x * blockDim.x + threadIdx.x;
  if (t >= total) return;
  float v = z[t] + b[t % cols];
  z[t] = v > 0.0f ? v : expm1f(v);
}

__global__ void pad_fc2_kernel(const float* w, float* bp) {
  int t = blockIdx.x * blockDim.x + threadIdx.x;
  if (t >= 128 * 16) return;
  int i = t >> 4, j = t & 15;
  bp[t] = (j < 10) ? w[i * 10 + j] : 0.0f;
}

__global__ void logsoftmax_out_kernel(const float* z2, const float* b, float* out) {
  int g = blockIdx.x * blockDim.x + threadIdx.x;
  if (g >= 64) return;
  float v[10];
  float mx = -1e30f;
#pragma unroll
  for (int j = 0; j < 10; ++j) {
    v[j] = z2[g * 16 + j] + b[j];
    mx = fmaxf(mx, v[j]);
  }
  float s = 0.0f;
#pragma unroll
  for (int j = 0; j < 10; ++j) s += expf(v[j] - mx);
  float l = mx + logf(s);
#pragma unroll
  for (int j = 0; j < 10; ++j) out[g * 10 + j] = v[j] - l;
}

// ---------------- launch ----------------

extern "C" void kernel_launch(void* const* d_in, const int* in_sizes, int n_in,
                              void* d_out, int out_size, void* d_ws, size_t ws_size,
                              hipStream_t stream) {
  (void)n_in; (void)out_size; (void)ws_size;
  const float* x    = (const float*)d_in[0];
  const float* pos  = (const float*)d_in[1];
  const int*   row  = (const int*)d_in[2];
  const int*   col  = (const int*)d_in[3];
  const float* W1   = (const float*)d_in[4];
  const float* R1   = (const float*)d_in[5];
  const float* W2   = (const float*)d_in[6];
  const float* R2   = (const float*)d_in[7];
  const float* W3   = (const float*)d_in[8];
  const float* R3   = (const float*)d_in[9];
  const float* fc1w = (const float*)d_in[10];
  const float* fc1b = (const float*)d_in[11];
  const float* fc2w = (const float*)d_in[12];
  const float* fc2b = (const float*)d_in[13];
  float* out = (float*)d_out;

  const int Bg   = 64;
  const int Ntot = in_sizes[0];     // 65536 nodes, layer 1
  const int E    = in_sizes[2];     // 524288 edges, layer 1
  const int C1   = Bg * 64;         // 4096 nodes after pool1 (G=8)
  const int C2   = Bg * 16;         // 1024 nodes after pool2 (G=4)
  const int nP2  = Bg * 64 * 64;    // 262144 dense pairs, layer 2
  const int nP3  = Bg * 16 * 16;    // 16384 dense pairs, layer 3

  float* ws = (float*)d_ws;
  size_t off = 0;
  auto take = [&](size_t n) -> size_t {
    size_t r = off;
    off += (n + 63) & ~(size_t)63;
    return r;
  };
  size_t o_m       = take(3);
  size_t o_agg1    = take((size_t)Ntot * 32);   // doubles as h1
  size_t o_deg1    = take(Ntot);
  size_t o_nocl1   = take(Ntot);                // int
  size_t o_feat1   = take((size_t)C1 * 32);
  size_t o_cnt1    = take(C1);
  size_t o_psum1   = take((size_t)C1 * 2);      // doubles as pos1
  size_t o_exist1  = take(nP2);                 // int
  size_t o_bm2     = take((size_t)32 * 1600);
  size_t o_xw2     = take((size_t)C1 * 1600);
  size_t o_root2   = take((size_t)C1 * 64);
  size_t o_agg2    = take((size_t)C1 * 64);     // doubles as h2
  size_t o_deg2    = take(C1);
  size_t o_nocl2   = take(C1);                  // int
  size_t o_feat2   = take((size_t)C2 * 64);
  size_t o_cnt2    = take(C2);
  size_t o_psum2   = take((size_t)C2 * 2);      // doubles as pos2
  size_t o_exist2  = take(nP3);                 // int
  size_t o_bm3     = take((size_t)64 * 1600);
  size_t o_xw3     = take((size_t)C2 * 1600);
  size_t o_root3   = take((size_t)C2 * 64);
  size_t o_agg3    = take((size_t)C2 * 64);     // doubles as h3
  size_t o_deg3    = take(C2);
  size_t o_feat4   = take((size_t)Bg * 4 * 64); // == z (64 x 256)
  size_t o_z1      = take((size_t)Bg * 128);
  size_t o_b2p     = take((size_t)128 * 16);
  size_t o_z2      = take((size_t)Bg * 16);
  size_t totalF = off;

  auto blk = [](long long thr) { return (unsigned)((thr + 255) / 256); };
  auto gemm = [&](const float* A, int lda, const float* Bm, int ldb, float* C, int ldc,
                  int M, int K, int N) {
    int ntiles = N / 16;
    int nfull = ntiles / 4;
    if (nfull > 0) {
      long long thr = (long long)(M / 16) * nfull * 32;
      wmma_gemm_kernel<4><<<blk(thr), 256, 0, stream>>>(A, lda, Bm, ldb, C, ldc, M, K,
                                                        nfull, 0);
    }
    int rem = ntiles % 4;
    if (rem > 0) {
      long long thr = (long long)(M / 16) * rem * 32;
      wmma_gemm_kernel<1><<<blk(thr), 256, 0, stream>>>(A, lda, Bm, ldb, C, ldc, M, K,
                                                        rem, nfull * 4);
    }
  };

  // 0) zero whole workspace (agg/deg/cnt/feat/exist/m all need 0 init)
  zero_kernel<<<4096, 256, 0, stream>>>(ws, totalF);

  // ---- layer 1 ----
  absmax_edges_kernel<<<blk(E), 256, 0, stream>>>(pos, row, col, E, ws + o_m);
  spline_edge_l1_kernel<<<blk((long long)E * 32), 256, 0, stream>>>(
      x, pos, row, col, W1, ws + o_m, ws + o_agg1, ws + o_deg1, E);
  node_fin_l1_kernel<<<blk((long long)Ntot * 32), 256, 0, stream>>>(
      ws + o_agg1, ws + o_deg1, x, R1, Ntot);

  // ---- pool 1 (G=8, P=64) ----
  pool_nodes_kernel<<<blk((long long)Ntot * 32), 256, 0, stream>>>(
      ws + o_agg1, 32, pos, 1024, 8, (int*)(ws + o_nocl1), ws + o_cnt1, ws + o_psum1,
      ws + o_feat1, Ntot);
  pool_div_kernel<<<blk(C1), 256, 0, stream>>>(ws + o_psum1, ws + o_cnt1, C1);
  pool_edges_l1_kernel<<<blk(E), 256, 0, stream>>>(
      row, col, (int*)(ws + o_nocl1), 64, (int*)(ws + o_exist1), E);

  // ---- layer 2 (F 32 -> 64) : XW via fp32 WMMA ----
  repack_w_kernel<<<blk(32 * 1600), 256, 0, stream>>>(W2, 32, ws + o_bm2, 32 * 1600);
  gemm(ws + o_feat1, 32, ws + o_bm2, 1600, ws + o_xw2, 1600, C1, 32, 1600);
  gemm(ws + o_feat1, 32, R2, 64, ws + o_root2, 64, C1, 32, 64);
  absmax_dense_kernel<<<blk(nP2), 256, 0, stream>>>(
      ws + o_psum1, (int*)(ws + o_exist1), 64, nP2, ws + o_m + 1);
  spline_edge_dense_kernel<<<blk((long long)nP2 * 32), 256, 0, stream>>>(
      ws + o_xw2, ws + o_psum1, (int*)(ws + o_exist1), ws + o_m + 1, 64, ws + o_agg2,
      ws + o_deg2, nP2);
  node_fin_kernel<<<blk((long long)C1 * 32), 256, 0, stream>>>(
      ws + o_agg2, ws + o_deg2, ws + o_root2, 64, C1);

  // ---- pool 2 (G=4, P=16) ----
  pool_nodes_kernel<<<blk((long long)C1 * 32), 256, 0, stream>>>(
      ws + o_agg2, 64, ws + o_psum1, 64, 4, (int*)(ws + o_nocl2), ws + o_cnt2,
      ws + o_psum2, ws + o_feat2, C1);
  pool_div_kernel<<<blk(C2), 256, 0, stream>>>(ws + o_psum2, ws + o_cnt2, C2);
  pool_edges_dense_kernel<<<blk(nP2), 256, 0, stream>>>(
      (int*)(ws + o_exist1), 64, (int*)(ws + o_nocl2), 16, (int*)(ws + o_exist2), nP2);

  // ---- layer 3 (F 64 -> 64) ----
  repack_w_kernel<<<blk(64 * 1600), 256, 0, stream>>>(W3, 64, ws + o_bm3, 64 * 1600);
  gemm(ws + o_feat2, 64, ws + o_bm3, 1600, ws + o_xw3, 1600, C2, 64, 1600);
  gemm(ws + o_feat2, 64, R3, 64, ws + o_root3, 64, C2, 64, 64);
  absmax_dense_kernel<<<blk(nP3), 256, 0, stream>>>(
      ws + o_psum2, (int*)(ws + o_exist2), 16, nP3, ws + o_m + 2);
  spline_edge_dense_kernel<<<blk((long long)nP3 * 32), 256, 0, stream>>>(
      ws + o_xw3, ws + o_psum2, (int*)(ws + o_exist2), ws + o_m + 2, 16, ws + o_agg3,
      ws + o_deg3, nP3);
  node_fin_kernel<<<blk((long long)C2 * 32), 256, 0, stream>>>(
      ws + o_agg3, ws + o_deg3, ws + o_root3, 64, C2);

  // ---- final 2x2 pool -> z (64 x 256), FC head ----
  final_pool_kernel<<<blk((long long)C2 * 32), 256, 0, stream>>>(
      ws + o_agg3, ws + o_psum2, C2, 16, ws + o_feat4);
  gemm(ws + o_feat4, 256, fc1w, 128, ws + o_z1, 128, Bg, 256, 128);
  elu_bias_kernel<<<blk(Bg * 128), 256, 0, stream>>>(ws + o_z1, fc1b, Bg * 128, 128);
  pad_fc2_kernel<<<blk(128 * 16), 256, 0, stream>>>(fc2w, ws + o_b2p);
  gemm(ws + o_z1, 128, ws + o_b2p, 16, ws + o_z2, 16, Bg, 128, 16);
  logsoftmax_out_kernel<<<1, 64, 0, stream>>>(ws + o_z2, fc2b, out);
}